// LimbDarkLightCurve_36670430773957
// MI455X (gfx1250) — compile-verified
//
#include <hip/hip_runtime.h>
#include <hip/hip_bf16.h>

// ---------------------------------------------------------------------------
// LimbDarkLightCurve for MI455X (gfx1250, wave32)
// Compute-bound elementwise kernel (~300 f32 ops/elem, dominated by
// acos/sqrt/rcp). Data path: async global->LDS double-buffered pipeline
// (gfx1250 GLOBAL_LOAD_ASYNC_TO_LDS_B128 + s_wait_asynccnt) when available.
// ---------------------------------------------------------------------------

#ifndef __has_builtin
#define __has_builtin(x) 0
#endif

#if defined(__HIP_DEVICE_COMPILE__)
#if __has_builtin(__builtin_amdgcn_global_load_async_to_lds_b128)
#define ASYNC_LDS 1
#else
#define ASYNC_LDS 0
#warning "gfx1250 async-to-LDS builtin NOT available; using direct global loads"
#endif
#else
#define ASYNC_LDS 0
#endif

#define BLOCK 256
#define VPT 4                    // float4 per thread
#define TILE (BLOCK * VPT)       // 1024 elements per tile
#define Q 10                     // Gauss-Legendre order (static in reference)

#if ASYNC_LDS
typedef int v4i __attribute__((vector_size(16)));
typedef __attribute__((address_space(1))) v4i Gv4i;   // global 128-bit chunk
typedef __attribute__((address_space(3))) v4i Lv4i;   // LDS 128-bit chunk
#endif

// Branchless acos, Cephes-style: |x|<=0.5 uses asin core directly;
// |x|>0.5 uses acos(x) = 2*asin(sqrt((1-x)/2)). ~2-3 ulp, plenty for the
// clipped argument range [-1+1e-7, 1-1e-7].
__device__ __forceinline__ float fast_acosf(float xx) {
    float ax  = fabsf(xx);
    bool  big = ax > 0.5f;
    float z   = big ? fmaf(-0.5f, ax, 0.5f) : xx * xx;
    float t   = big ? __builtin_amdgcn_sqrtf(z) : ax;
    float p   = fmaf(z, 4.2163199048e-2f, 2.4181311049e-2f);
    p         = fmaf(z, p, 4.5470025998e-2f);
    p         = fmaf(z, p, 7.4953002686e-2f);
    p         = fmaf(z, p, 1.6666752422e-1f);
    float r   = fmaf(p * z, t, t);          // asin(t)
    float pos = big ? (r + r)                : (1.5707963267948966f - r);
    float neg = big ? (3.14159265358979f - (r + r)) : (1.5707963267948966f + r);
    return xx < 0.0f ? neg : pos;
}

__device__ __forceinline__ float transit_elem(float px, float py, float pz,
                                              float u0, float u1, float r,
                                              float inv_norm) {
    // leggauss(10) nodes and 2*weights (static constants -> folded literals)
    const float nodes[Q] = {
        -0.9739065285171717f, -0.8650633666889845f, -0.6794095682990244f,
        -0.4333953941292472f, -0.1488743389816312f,  0.1488743389816312f,
         0.4333953941292472f,  0.6794095682990244f,  0.8650633666889845f,
         0.9739065285171717f };
    const float w2[Q] = {
        0.1333426886173762f, 0.2989026983011612f, 0.4381727250319640f,
        0.5385334386199926f, 0.5910484494295058f, 0.5910484494295058f,
        0.5385334386199926f, 0.4381727250319640f, 0.2989026983011612f,
        0.1333426886173762f };

    float b2   = fmaf(px, px, py * py);
    float b    = __builtin_amdgcn_sqrtf(b2);
    float lo   = fmaxf(b - r, 0.0f);
    float hi   = fminf(b + r, 1.0f);
    float hw   = 0.5f * fmaxf(hi - lo, 0.0f);   // zero width => no occultation
    float mid  = 0.5f * (hi + lo);
    float bmr  = fmaf(-r, r, b2);               // b^2 - r^2
    float twob = b + b;
    float acc  = 0.0f;
#pragma unroll
    for (int q = 0; q < Q; ++q) {
        float rho   = fmaf(hw, nodes[q], mid);
        float rho2  = rho * rho;
        float mu    = __builtin_amdgcn_sqrtf(fmaxf(1.0f - rho2, 0.0f));
        float omm   = 1.0f - mu;
        float prof  = fmaf(-omm, fmaf(u1, omm, u0), 1.0f);  // 1 - u0*omm - u1*omm^2
        float denom = fmaxf(twob * rho, 1e-12f);
        float carg  = (rho2 + bmr) * __builtin_amdgcn_rcpf(denom);
        carg        = fminf(fmaxf(carg, -0.9999999f), 0.9999999f);
        float phi   = fast_acosf(carg);
        acc         = fmaf(w2[q] * rho, phi * prof, acc);
    }
    float lc = (hw * acc) * inv_norm;           // inv_norm = -1/norm
    return pz > 0.0f ? lc : 0.0f;
}

__global__ __launch_bounds__(BLOCK) void limbdark_kernel(
    const float* __restrict__ x, const float* __restrict__ y,
    const float* __restrict__ zl, const float* __restrict__ u,
    const float* __restrict__ rp, float* __restrict__ out,
    int n, int ntiles)
{
#if ASYNC_LDS
    __shared__ float lds[2][3][TILE];           // 24 KB double-buffered x/y/z
#endif
    const int tid  = threadIdx.x;
    const int toff = tid * VPT;

    const float u0 = u[0], u1 = u[1], r = rp[0];
    // norm = pi*(1 - u0*2/(2*3) - u1*2/(3*4))
    const float norm     = 3.14159265358979f *
                           (1.0f - u0 * (1.0f / 3.0f) - u1 * (1.0f / 6.0f));
    const float inv_norm = -1.0f / norm;

    long tile          = blockIdx.x;
    const long tstride = gridDim.x;

#if ASYNC_LDS
    int buf = 0;
    {   // prologue: prefetch first tile into buf 0
        const long e0 = tile * (long)TILE + toff;
        if (e0 + 3 < (long)n) {
            __builtin_amdgcn_global_load_async_to_lds_b128(
                (Gv4i*)(const void*)(x + e0),  (Lv4i*)&lds[0][0][toff], 0, 0);
            __builtin_amdgcn_global_load_async_to_lds_b128(
                (Gv4i*)(const void*)(y + e0),  (Lv4i*)&lds[0][1][toff], 0, 0);
            __builtin_amdgcn_global_load_async_to_lds_b128(
                (Gv4i*)(const void*)(zl + e0), (Lv4i*)&lds[0][2][toff], 0, 0);
        }
    }
#endif

    for (; tile < (long)ntiles; tile += tstride) {
        const long e0  = tile * (long)TILE + toff;
        const bool vec = (e0 + 3 < (long)n);
        float4 xv = make_float4(0.f, 0.f, 0.f, 0.f);
        float4 yv = xv, zv = xv;

#if ASYNC_LDS
        // current tile's async copies are the oldest outstanding -> drain all
        asm volatile("s_wait_asynccnt 0x0" ::: "memory");
        if (vec) {
            xv = *reinterpret_cast<const float4*>(&lds[buf][0][toff]);
            yv = *reinterpret_cast<const float4*>(&lds[buf][1][toff]);
            zv = *reinterpret_cast<const float4*>(&lds[buf][2][toff]);
        }
        // ensure ds reads landed in VGPRs before the async engine may
        // overwrite this buffer in the next iteration
        asm volatile("s_wait_dscnt 0x0" ::: "memory");
        {   // prefetch next tile into the other buffer (overlaps compute)
            const long nt = tile + tstride;
            if (nt < (long)ntiles) {
                const long ne0 = nt * (long)TILE + toff;
                if (ne0 + 3 < (long)n) {
                    const int nb = buf ^ 1;
                    __builtin_amdgcn_global_load_async_to_lds_b128(
                        (Gv4i*)(const void*)(x + ne0),  (Lv4i*)&lds[nb][0][toff], 0, 0);
                    __builtin_amdgcn_global_load_async_to_lds_b128(
                        (Gv4i*)(const void*)(y + ne0),  (Lv4i*)&lds[nb][1][toff], 0, 0);
                    __builtin_amdgcn_global_load_async_to_lds_b128(
                        (Gv4i*)(const void*)(zl + ne0), (Lv4i*)&lds[nb][2][toff], 0, 0);
                }
            }
        }
#else
        if (vec) {
            xv = *reinterpret_cast<const float4*>(x + e0);
            yv = *reinterpret_cast<const float4*>(y + e0);
            zv = *reinterpret_cast<const float4*>(zl + e0);
        }
#endif

        if (vec) {
            float4 res;
            res.x = transit_elem(xv.x, yv.x, zv.x, u0, u1, r, inv_norm);
            res.y = transit_elem(xv.y, yv.y, zv.y, u0, u1, r, inv_norm);
            res.z = transit_elem(xv.z, yv.z, zv.z, u0, u1, r, inv_norm);
            res.w = transit_elem(xv.w, yv.w, zv.w, u0, u1, r, inv_norm);
            *reinterpret_cast<float4*>(out + e0) = res;
        } else {
            // ragged tail: scalar, direct from global
            for (int k = 0; k < VPT; ++k) {
                const long e = e0 + k;
                if (e < (long)n)
                    out[e] = transit_elem(x[e], y[e], zl[e], u0, u1, r, inv_norm);
            }
        }
#if ASYNC_LDS
        buf ^= 1;
#endif
    }
}

extern "C" void kernel_launch(void* const* d_in, const int* in_sizes, int n_in,
                              void* d_out, int out_size, void* d_ws, size_t ws_size,
                              hipStream_t stream) {
    (void)n_in; (void)d_ws; (void)ws_size; (void)out_size;
    const float* x  = (const float*)d_in[0];
    const float* y  = (const float*)d_in[1];
    const float* z  = (const float*)d_in[2];
    const float* u  = (const float*)d_in[3];   // 2 limb-darkening coeffs
    const float* r  = (const float*)d_in[4];   // radius ratio (1 elem)
    // d_in[5] = ld_order (int, ==10); quadrature table is hardcoded to match
    float* out = (float*)d_out;

    const int n      = in_sizes[0];
    const int ntiles = (n + TILE - 1) / TILE;
    int grid         = ntiles < 2048 ? ntiles : 2048;
    if (grid < 1) grid = 1;

    limbdark_kernel<<<grid, BLOCK, 0, stream>>>(x, y, z, u, r, out, n, ntiles);
}